// ColorHistograms_30846455120255
// MI455X (gfx1250) — compile-verified
//
#include <hip/hip_runtime.h>
#include <hip/hip_bf16.h>
#include <math.h>

typedef __attribute__((ext_vector_type(2))) float v2f;
typedef __attribute__((ext_vector_type(8))) float v8f;

#define NUM_BINS   512
#define PIX        4096      // 64*64
#define D          512
#define T_DIM      256
#define B_DIM      16
#define NOUT       128
#define WIN        101
#define HALF       50

__device__ __forceinline__ int bin_of(int r, int g, int b) {
    return ((r >> 5) << 6) | ((g >> 5) << 3) | (b >> 5);
}

// ---------------------------------------------------------------------------
// Kernel 1: per-frame 512-bin color histogram + L2 normalization.
// One block (256 threads) per frame; hist in LDS via ds_add atomics.
// Bandwidth-bound stage: each thread consumes 4 pixels (48 B) per step via
// three b128 loads so the wave issues wide contiguous VMEM requests.
// ---------------------------------------------------------------------------
__global__ __launch_bounds__(256)
void hist_kernel(const int* __restrict__ frames, float* __restrict__ x) {
    __shared__ int   hist[NUM_BINS];
    __shared__ float ssum;
    const int f   = blockIdx.x;          // 0..4095 frame id
    const int tid = threadIdx.x;         // 0..255

    hist[tid]       = 0;
    hist[tid + 256] = 0;
    if (tid == 0) ssum = 0.0f;
    __syncthreads();

    const int4* fp4 = (const int4*)(frames + (size_t)f * (PIX * 3));
    // 4096 pixels = 1024 chunks of 4 pixels (3 x int4); 256 threads -> 4 iters
    #pragma unroll
    for (int it = 0; it < 4; ++it) {
        const int c = it * 256 + tid;        // chunk id, contiguous per wave
        const int4 a = fp4[3 * c + 0];
        const int4 b = fp4[3 * c + 1];
        const int4 d = fp4[3 * c + 2];
        atomicAdd(&hist[bin_of(a.x, a.y, a.z)], 1);
        atomicAdd(&hist[bin_of(a.w, b.x, b.y)], 1);
        atomicAdd(&hist[bin_of(b.z, b.w, d.x)], 1);
        atomicAdd(&hist[bin_of(d.y, d.z, d.w)], 1);
    }
    __syncthreads();

    const float h0 = (float)hist[tid];
    const float h1 = (float)hist[tid + 256];
    atomicAdd(&ssum, h0 * h0 + h1 * h1);
    __syncthreads();

    const float scale = 1.0f / fmaxf(sqrtf(ssum), 1e-12f);
    float* xp = x + (size_t)f * D;
    xp[tid]       = h0 * scale;
    xp[tid + 256] = h1 * scale;
}

// ---------------------------------------------------------------------------
// Kernel 2: sim[b] = x[b] (256x512) @ x[b]^T  -> (256x256), exact f32 WMMA.
// One wave per 16x16 output tile; 4 waves per block.
// grid = (4, 16, 16) = (n-tile/4, m-tile, batch), block = 128.
// ---------------------------------------------------------------------------
__global__ __launch_bounds__(128)
void sim_kernel(const float* __restrict__ x, float* __restrict__ sim) {
    const int wave = threadIdx.x >> 5;
    const int lane = threadIdx.x & 31;
    const int tn   = blockIdx.x * 4 + wave;   // 0..15
    const int tm   = blockIdx.y;              // 0..15
    const int b    = blockIdx.z;              // 0..15

    const float* xb = x + (size_t)b * T_DIM * D;
    const int mn    = lane & 15;              // row for A, col for B
    const int khalf = (lane >> 4) * 2;        // 0 or 2

    const float* arow = xb + (size_t)(tm * 16 + mn) * D;
    const float* brow = xb + (size_t)(tn * 16 + mn) * D;   // B = x^T: B(k,n)=x[n*D+k]

    v8f c = {};
    #pragma unroll 8
    for (int k0 = 0; k0 < D; k0 += 4) {
        v2f a  = *(const v2f*)(arow + k0 + khalf);
        v2f bb = *(const v2f*)(brow + k0 + khalf);
        c = __builtin_amdgcn_wmma_f32_16x16x4_f32(false, a, false, bb,
                                                  (short)0, c, false, false);
    }

    float* sp = sim + (size_t)b * T_DIM * T_DIM
                    + (size_t)(tm * 16) * T_DIM + tn * 16;
    const int mbase = (lane >> 4) * 8;
    #pragma unroll
    for (int i = 0; i < 8; i++) {
        sp[(size_t)(mbase + i) * T_DIM + mn] = c[i];
    }
}

// ---------------------------------------------------------------------------
// Kernel 3: out[b,t,o] = relu( sum_w sim[b, t, t+w-50] * fc_w[o,w] + fc_b[o] )
// Band-gather fused into the A-operand load; K padded 101 -> 104.
// One wave per 16x16 output tile. grid = (2, 16, 16), block = 128.
// ---------------------------------------------------------------------------
__global__ __launch_bounds__(128)
void fc_kernel(const float* __restrict__ sim, const float* __restrict__ fc_w,
               const float* __restrict__ fc_b, float* __restrict__ out) {
    const int wave = threadIdx.x >> 5;
    const int lane = threadIdx.x & 31;
    const int tn   = blockIdx.x * 4 + wave;   // 0..7  (output-col tile)
    const int tt   = blockIdx.y;              // 0..15 (t tile)
    const int b    = blockIdx.z;              // 0..15

    const int mn    = lane & 15;
    const int khalf = (lane >> 4) * 2;
    const int t     = tt * 16 + mn;           // A row: time index
    const int o     = tn * 16 + mn;           // B col: output channel

    const float* simrow = sim + (size_t)b * T_DIM * T_DIM + (size_t)t * T_DIM;
    const float* wrow   = fc_w + (size_t)o * WIN;

    v8f c = {};
    for (int k0 = 0; k0 < 104; k0 += 4) {
        const int w0 = k0 + khalf;
        const int w1 = w0 + 1;
        const int s0 = t + w0 - HALF;
        const int s1 = t + w1 - HALF;
        v2f a, bb;
        a.x  = (w0 < WIN && s0 >= 0 && s0 < T_DIM) ? simrow[s0] : 0.0f;
        a.y  = (w1 < WIN && s1 >= 0 && s1 < T_DIM) ? simrow[s1] : 0.0f;
        bb.x = (w0 < WIN) ? wrow[w0] : 0.0f;
        bb.y = (w1 < WIN) ? wrow[w1] : 0.0f;
        c = __builtin_amdgcn_wmma_f32_16x16x4_f32(false, a, false, bb,
                                                  (short)0, c, false, false);
    }

    const float bias  = fc_b[o];
    const int   mbase = (lane >> 4) * 8;
    float* op = out + ((size_t)b * T_DIM + tt * 16) * NOUT + tn * 16;
    #pragma unroll
    for (int i = 0; i < 8; i++) {
        op[(size_t)(mbase + i) * NOUT + mn] = fmaxf(c[i] + bias, 0.0f);
    }
}

// ---------------------------------------------------------------------------
extern "C" void kernel_launch(void* const* d_in, const int* in_sizes, int n_in,
                              void* d_out, int out_size, void* d_ws, size_t ws_size,
                              hipStream_t stream) {
    const int*   frames = (const int*)d_in[0];   // (16,256,64,64,3) int32
    const float* fc_w   = (const float*)d_in[1]; // (128,101)
    const float* fc_b   = (const float*)d_in[2]; // (128,)
    float*       out    = (float*)d_out;         // (16,256,128)

    // workspace layout: x (4096*512 f32 = 8MB) | sim (16*256*256 f32 = 4MB)
    float* x_ws   = (float*)d_ws;
    float* sim_ws = x_ws + (size_t)B_DIM * T_DIM * D;

    hist_kernel<<<dim3(B_DIM * T_DIM), dim3(256), 0, stream>>>(frames, x_ws);
    sim_kernel<<<dim3(4, 16, 16), dim3(128), 0, stream>>>(x_ws, sim_ws);
    fc_kernel<<<dim3(2, 16, 16), dim3(128), 0, stream>>>(sim_ws, fc_w, fc_b, out);
}